// GHMCLoss_13846974562932
// MI455X (gfx1250) — compile-verified
//
#include <hip/hip_runtime.h>
#include <stdint.h>

// GHM-C loss forward for MI455X (gfx1250, wave32).
// Memory-bound: ~1.17 GB total traffic -> ~50us at 23.3 TB/s.
//   Pass 1: histogram + loss partials + cache 8-bit bins   (read 671MB, write 84MB)
//   Pass 2: EMA update + recip table + deterministic loss  (tiny)
//   Pass 3: weights via LDS table lookup                   (read 84MB, write 336MB)

#define GH_BINS 100
#define GH_C    40
#define GH_NBIN (GH_C * GH_BINS)   // 4000
#define GH_B    2097152
#define NB1     4096               // blocks in main pass
#define TPB     256                // 8 waves (wave32) per block

// ---------------------------------------------------------------- pass 0: zero counts
__global__ void ghmc_init(unsigned* __restrict__ counts) {
    int i = blockIdx.x * blockDim.x + threadIdx.x;
    if (i < GH_NBIN) counts[i] = 0u;
}

// ---------------------------------------------------------------- pass 1: histogram + loss + bins
__global__ void ghmc_main(const float4* __restrict__ pred4,
                          const float4* __restrict__ tgt4,
                          unsigned* __restrict__ counts,
                          float* __restrict__ partial,
                          uint8_t* __restrict__ bins,   // may be null
                          int nv) {
    __shared__ unsigned hist[GH_NBIN];   // 16 KB per-block histogram
    __shared__ float    red[TPB];

    for (int k = threadIdx.x; k < GH_NBIN; k += TPB) hist[k] = 0u;
    __syncthreads();

    const int stride = gridDim.x * TPB;
    float lacc = 0.0f;

    for (int i = blockIdx.x * TPB + threadIdx.x; i < nv; i += stride) {
        // speculative prefetch of next tile (-> global_prefetch_b8)
        __builtin_prefetch(pred4 + i + stride, 0, 1);
        __builtin_prefetch(tgt4  + i + stride, 0, 1);

        float4 p = pred4[i];
        float4 t = tgt4[i];
        const int base = i * 4;
        float pv[4] = {p.x, p.y, p.z, p.w};
        float tv[4] = {t.x, t.y, t.z, t.w};
        uint32_t packed = 0u;

        #pragma unroll
        for (int j = 0; j < 4; ++j) {
            float pp = pv[j], tt = tv[j];
            float s  = 1.0f / (1.0f + expf(-pp));          // sigmoid
            float g  = fabsf(s - tt);
            int   b  = (int)(g * 100.0f);
            b = b < 0 ? 0 : (b > GH_BINS - 1 ? GH_BINS - 1 : b);
            int   c  = (base + j) % GH_C;
            atomicAdd(&hist[c * GH_BINS + b], 1u);         // ds_add_u32
            // BCE-with-logits term
            lacc += fmaxf(pp, 0.0f) - pp * tt + log1pf(expf(-fabsf(pp)));
            packed |= ((uint32_t)b) << (8 * j);
        }
        if (bins) ((uint32_t*)bins)[i] = packed;           // 4 bins per b32 store
    }

    __syncthreads();
    for (int k = threadIdx.x; k < GH_NBIN; k += TPB) {
        unsigned h = hist[k];
        if (h) atomicAdd(&counts[k], h);                   // integer atomics: deterministic
    }

    // deterministic in-block tree reduction of loss
    red[threadIdx.x] = lacc;
    __syncthreads();
    for (int off = TPB / 2; off > 0; off >>= 1) {
        if (threadIdx.x < off) red[threadIdx.x] += red[threadIdx.x + off];
        __syncthreads();
    }
    if (threadIdx.x == 0) partial[blockIdx.x] = red[0];
}

// ---------------------------------------------------------------- pass 2: EMA + recip + loss
__global__ void ghmc_finalize(const unsigned* __restrict__ counts,
                              const float* __restrict__ acc_sum,
                              const float* __restrict__ partial,
                              float* __restrict__ out,        // full d_out
                              float* __restrict__ recip,
                              int n_total, int nblocks1) {
    int gid = blockIdx.x * blockDim.x + threadIdx.x;
    if (gid < GH_NBIN) {
        unsigned cnt = counts[gid];
        float a  = acc_sum[gid];
        // match JAX: MMT and (1-MMT) are python doubles cast to f32 -> 0.6f and 0.4f
        float an = cnt ? (0.6f * a + 0.4f * (float)cnt) : a;
        out[1 + (size_t)n_total + gid] = an;                // acc_new output
        recip[gid] = (float)GH_B / an;                      // = weights value for (c,bin)
    }
    if (blockIdx.x == 0) {
        __shared__ float red[TPB];
        float s = 0.0f;
        for (int j = threadIdx.x; j < nblocks1; j += TPB) s += partial[j]; // fixed order
        red[threadIdx.x] = s;
        __syncthreads();
        for (int off = TPB / 2; off > 0; off >>= 1) {
            if (threadIdx.x < off) red[threadIdx.x] += red[threadIdx.x + off];
            __syncthreads();
        }
        if (threadIdx.x == 0) out[0] = red[0] / (float)n_total;  // loss
    }
}

// ---------------------------------------------------------------- LDS table broadcast (async DMA)
__device__ __forceinline__ void load_table_async(float* lds_tab, const float* __restrict__ gtab) {
    // 4000 floats = 1000 x b128 async global->LDS copies, tracked by ASYNCcnt.
    const int NQ = GH_NBIN / 4;
    for (int k = threadIdx.x; k < NQ; k += blockDim.x) {
        unsigned lds_addr = (unsigned)(uintptr_t)(&lds_tab[4 * k]); // low 32b of flat = LDS offset
        const float* gp = gtab + 4 * k;
        asm volatile("global_load_async_to_lds_b128 %0, %1, off"
                     :: "v"(lds_addr), "v"(gp) : "memory");
    }
    asm volatile("s_wait_asynccnt 0x0" ::: "memory");
    __syncthreads();
}

// ---------------------------------------------------------------- pass 3a: weights from cached bins
__global__ void ghmc_weights_bins(const uint32_t* __restrict__ bins4,
                                  const float* __restrict__ recip,
                                  float* __restrict__ wout,   // = d_out + 1
                                  int nv) {
    __shared__ float lr[GH_NBIN];
    load_table_async(lr, recip);

    const int stride = gridDim.x * blockDim.x;
    for (int i = blockIdx.x * blockDim.x + threadIdx.x; i < nv; i += stride) {
        uint32_t pk = bins4[i];
        const int base = 4 * i;
        #pragma unroll
        for (int j = 0; j < 4; ++j) {
            int b = (pk >> (8 * j)) & 0xFF;
            int c = (base + j) % GH_C;
            wout[base + j] = lr[c * GH_BINS + b];
        }
    }
}

// ---------------------------------------------------------------- pass 3b: fallback (recompute bins)
__global__ void ghmc_weights_recompute(const float4* __restrict__ pred4,
                                       const float4* __restrict__ tgt4,
                                       const float* __restrict__ recip,
                                       float* __restrict__ wout,
                                       int nv) {
    __shared__ float lr[GH_NBIN];
    load_table_async(lr, recip);

    const int stride = gridDim.x * blockDim.x;
    for (int i = blockIdx.x * blockDim.x + threadIdx.x; i < nv; i += stride) {
        float4 p = pred4[i];
        float4 t = tgt4[i];
        const int base = 4 * i;
        float pv[4] = {p.x, p.y, p.z, p.w};
        float tv[4] = {t.x, t.y, t.z, t.w};
        #pragma unroll
        for (int j = 0; j < 4; ++j) {
            float s = 1.0f / (1.0f + expf(-pv[j]));
            float g = fabsf(s - tv[j]);
            int   b = (int)(g * 100.0f);
            b = b < 0 ? 0 : (b > GH_BINS - 1 ? GH_BINS - 1 : b);
            int   c = (base + j) % GH_C;
            wout[base + j] = lr[c * GH_BINS + b];
        }
    }
}

// ---------------------------------------------------------------- host launcher
extern "C" void kernel_launch(void* const* d_in, const int* in_sizes, int n_in,
                              void* d_out, int out_size, void* d_ws, size_t ws_size,
                              hipStream_t stream) {
    const float* pred = (const float*)d_in[0];
    const float* tgt  = (const float*)d_in[1];
    const float* accs = (const float*)d_in[2];
    float* out = (float*)d_out;

    const int N  = GH_B * GH_C;   // 83,886,080
    const int NV = N / 4;         // float4 count

    // workspace layout (all 16KB-block aligned regions first, bins last)
    uint8_t*  ws      = (uint8_t*)d_ws;
    unsigned* counts  = (unsigned*)(ws);            // 16000 B
    float*    partial = (float*)(ws + 16384);       // NB1*4 = 16384 B
    float*    recip   = (float*)(ws + 32768);       // 16000 B
    uint8_t*  bins    = ws + 49152;                 // N bytes
    const size_t need = 49152 + (size_t)N;
    const bool use_bins = (ws_size >= need);

    ghmc_init<<<(GH_NBIN + TPB - 1) / TPB, TPB, 0, stream>>>(counts);

    ghmc_main<<<NB1, TPB, 0, stream>>>((const float4*)pred, (const float4*)tgt,
                                       counts, partial,
                                       use_bins ? bins : (uint8_t*)nullptr, NV);

    ghmc_finalize<<<16, TPB, 0, stream>>>(counts, accs, partial, out, recip, N, NB1);

    if (use_bins) {
        ghmc_weights_bins<<<8192, TPB, 0, stream>>>((const uint32_t*)bins, recip, out + 1, NV);
    } else {
        ghmc_weights_recompute<<<8192, TPB, 0, stream>>>((const float4*)pred, (const float4*)tgt,
                                                         recip, out + 1, NV);
    }
}